// BSplineActivation_67800353734729
// MI455X (gfx1250) — compile-verified
//
#include <hip/hip_runtime.h>
#include <hip/hip_bf16.h>
#include <stdint.h>

// B-spline activation, piecewise-cubic rewrite.
//   Pass 1 (tiny): per (feature, interval) exact cubic coefficients about the
//                  center m_i = t0 + (i+0.5)*step, scaler folded in.
//                  768*10 float4 entries -> d_ws (120 KB).
//   Pass 2 (streaming): interval lookup + 3 FMAs per element; table staged in
//                  LDS via gfx1250 global_load_async_to_lds_b128 (ASYNCcnt).

#define NF 768
#define NG 5
#define NI 10          // intervals in extended knot vector (11 knots)
#define EPSV 1e-8f
#define FT 256         // features per block tile
#define NROWS 16384    // B*S

__global__ __launch_bounds__(256)
void spline_build_table(const float* __restrict__ knots,
                        const float* __restrict__ coeffs,
                        const float* __restrict__ scaler,
                        float4* __restrict__ tabA) {  // [NI][NF] : a0..a3
    int g = blockIdx.x * blockDim.x + threadIdx.x;
    if (g >= NF * NI) return;
    const int f = g % NF;
    const int i = g / NF;

    // extended knots: 3 padded each side, uniform step as in reference
    float t[11];
    const float k0 = knots[f * NG + 0];
    const float k4 = knots[f * NG + NG - 1];
    const float step = (k4 - k0) * 0.25f;
#pragma unroll
    for (int j = 0; j < NG; ++j) t[3 + j] = knots[f * NG + j];
    t[0] = k0 - 3.f * step; t[1] = k0 - 2.f * step; t[2] = k0 - step;
    t[8] = k4 + step;       t[9] = k4 + 2.f * step; t[10] = k4 + 3.f * step;

    // polynomial center: MUST match spline_eval's expression bit-for-bit.
    const float t0v = k0 - 3.f * step;
    const float m = fmaf((float)i + 0.5f, step, t0v);

    // polynomial Cox-de Boor restricted to interval i, variable dx = x - m.
    // Q[jp] = coeffs (order 0..3) of basis B_{i-3+jp}^d on this interval.
    float Q[4][4];
#pragma unroll
    for (int a = 0; a < 4; ++a)
#pragma unroll
        for (int c = 0; c < 4; ++c) Q[a][c] = 0.f;
    Q[3][0] = 1.0f;  // degree 0: indicator of interval i

#pragma unroll
    for (int d = 1; d <= 3; ++d) {
        float R[4][4];
#pragma unroll
        for (int a = 0; a < 4; ++a)
#pragma unroll
            for (int c = 0; c < 4; ++c) R[a][c] = 0.f;
#pragma unroll
        for (int jp = 0; jp < 4; ++jp) {
            const int j = i - 3 + jp;
            if (j < 0 || j > 9 - d) continue;  // basis doesn't exist
            const float r1 = 1.0f / (t[j + d]     - t[j]     + EPSV);
            const float r2 = 1.0f / (t[j + d + 1] - t[j + 1] + EPSV);
            const float a1 = (m - t[j]) * r1,          b1 = r1;   // (x-t_j)*r1
            const float a2 = (t[j + d + 1] - m) * r2,  b2 = -r2;  // (t_{j+d+1}-x)*r2
#pragma unroll
            for (int c = 0; c < 4; ++c) {
                float v = a1 * Q[jp][c];
                if (c > 0) v += b1 * Q[jp][c - 1];
                if (jp < 3) {
                    v += a2 * Q[jp + 1][c];
                    if (c > 0) v += b2 * Q[jp + 1][c - 1];
                }
                R[jp][c] = v;
            }
        }
#pragma unroll
        for (int a = 0; a < 4; ++a)
#pragma unroll
            for (int c = 0; c < 4; ++c) Q[a][c] = R[a][c];
    }

    // combine with spline coefficients (j = 0..6) and scaler
    const float s = scaler[f];
    float p[4] = {0.f, 0.f, 0.f, 0.f};
#pragma unroll
    for (int jp = 0; jp < 4; ++jp) {
        const int j = i - 3 + jp;
        if (j < 0 || j > 6) continue;
        const float cj = coeffs[f * 7 + j] * s;
#pragma unroll
        for (int c = 0; c < 4; ++c) p[c] += cj * Q[jp][c];
    }
    tabA[g] = make_float4(p[0], p[1], p[2], p[3]);
}

__global__ __launch_bounds__(256)
void spline_eval(const float* __restrict__ x,
                 const float* __restrict__ knots,
                 const float4* __restrict__ tabA,
                 float* __restrict__ out,
                 int rowsPerBlock) {
    __shared__ float4 sA[NI * FT];   // 40 KB, layout [interval][featureLocal]
    const int tid = threadIdx.x;
    const int ft  = blockIdx.x;      // 0..2
    const int f   = ft * FT + tid;

    // ---- async fill of the per-tile table slice (gfx1250 ASYNCcnt path) ----
    {
        const uint64_t baseA = (uint64_t)(uintptr_t)(const void*)(tabA + ft * FT);
#pragma unroll
        for (int i = 0; i < NI; ++i) {
            unsigned goff = (unsigned)((i * NF + tid) * 16);
            unsigned lds  = (unsigned)(uintptr_t)(void*)&sA[i * FT + tid];
            asm volatile("global_load_async_to_lds_b128 %0, %1, %2"
                         :: "v"(lds), "v"(goff), "s"(baseA) : "memory");
        }
    }

    // ---- extended knots -> registers (overlaps the async copies) ----
    const float k0 = knots[f * NG + 0];
    const float ka = knots[f * NG + 1];
    const float kb = knots[f * NG + 2];
    const float kc = knots[f * NG + 3];
    const float k4 = knots[f * NG + 4];
    const float step = (k4 - k0) * 0.25f;
    const float t0 = k0 - 3.f * step, t1 = k0 - 2.f * step, t2 = k0 - step;
    const float t3 = k0, t4 = ka, t5 = kb, t6 = kc, t7 = k4;
    const float t8 = k4 + step, t9 = k4 + 2.f * step, t10 = k4 + 3.f * step;

    asm volatile("s_wait_asynccnt 0" ::: "memory");
    __syncthreads();

    // ---- streaming evaluation: one feature per lane, rows in sequence ----
    const int rowBase = blockIdx.y * rowsPerBlock;
#pragma unroll 8
    for (int r = 0; r < rowsPerBlock; ++r) {
        const int n = rowBase + r;
        const float xv = __builtin_nontemporal_load(&x[n * NF + f]);
        int idx = (xv >= t1) + (xv >= t2) + (xv >= t3) + (xv >= t4) + (xv >= t5)
                + (xv >= t6) + (xv >= t7) + (xv >= t8) + (xv >= t9);
        const bool valid = (xv >= t0) && (xv < t10);
        const float4 a = sA[idx * FT + tid];
        // center: identical expression to spline_build_table
        const float mm = fmaf((float)idx + 0.5f, step, t0);
        const float dx = xv - mm;
        float y = fmaf(fmaf(fmaf(a.w, dx, a.z), dx, a.y), dx, a.x);
        y = valid ? y : 0.0f;
        __builtin_nontemporal_store(y, &out[n * NF + f]);
    }
}

extern "C" void kernel_launch(void* const* d_in, const int* in_sizes, int n_in,
                              void* d_out, int out_size, void* d_ws, size_t ws_size,
                              hipStream_t stream) {
    const float* x      = (const float*)d_in[0];
    const float* knots  = (const float*)d_in[1];
    const float* coeffs = (const float*)d_in[2];
    const float* scaler = (const float*)d_in[3];
    float* out = (float*)d_out;

    // scratch layout: tabA = NI*NF float4 = 120 KB
    float4* tabA = (float4*)d_ws;

    spline_build_table<<<(NF * NI + 255) / 256, 256, 0, stream>>>(
        knots, coeffs, scaler, tabA);

    const int rowBlocks = 256;                  // 3 x 256 = 768 blocks
    const int rowsPerBlock = NROWS / rowBlocks; // 64 rows each
    dim3 grid(NF / FT, rowBlocks);
    spline_eval<<<grid, 256, 0, stream>>>(x, knots, tabA, out, rowsPerBlock);
}